// AscendRejectionSampler_19207093747782
// MI455X (gfx1250) — compile-verified
//
#include <hip/hip_runtime.h>
#include <cfloat>
#include <cstdint>
#include <cstddef>

// ---------------------------------------------------------------------------
// AscendRejectionSampler for MI455X (gfx1250).
// Bandwidth-bound dual-argmax row scan (no matmul -> no WMMA by design),
// using 128-bit NT streaming loads, global_prefetch_b8, and the gfx1250
// async-to-LDS path (ASYNCcnt) as a row-head warm-up.
// ---------------------------------------------------------------------------

typedef float f4 __attribute__((ext_vector_type(4)));

#define TPB 256

__global__ __launch_bounds__(TPB) void rs_row_scan(
    const float* __restrict__ target_probs,  // (N,V)
    const float* __restrict__ draft_probs,   // (N,V)
    const float* __restrict__ q,             // (B,V)
    const int*   __restrict__ cu,            // (B)
    const int*   __restrict__ draft_tok,     // (N)
    int*   __restrict__ tgt_am,              // (N) out
    int*   __restrict__ recovered,           // (N) out
    float* __restrict__ dp_out,              // (N) out
    float* __restrict__ tp_out,              // (N) out
    int B, int N, int V)
{
    const int row = blockIdx.x;
    if (row >= N) return;
    const int tid = threadIdx.x;

    __shared__ int   s_req;
    __shared__ float sTv[TPB];
    __shared__ int   sTi[TPB];
    __shared__ float sRv[TPB];
    __shared__ int   sRi[TPB];
    __shared__ __align__(16) unsigned char s_probe[16];

    const float* trow = target_probs + (size_t)row * (size_t)V;
    const float* drow = draft_probs  + (size_t)row * (size_t)V;

    if (tid == 0) {
        // req_id = searchsorted(cu, row, side='right')
        int r = B - 1;
        for (int b = 0; b < B; ++b) { if (cu[b] > row) { r = b; break; } }
        s_req = r;
        // scalar gathers at the draft token
        int tok = draft_tok[row];
        dp_out[row] = drow[tok];
        tp_out[row] = trow[tok];
        // CDNA5 async-to-LDS warm-up of the row head (exercises ASYNCcnt path)
        unsigned lds_off = (unsigned)(uintptr_t)&s_probe[0];
        unsigned long long ga = (unsigned long long)(uintptr_t)trow;
        asm volatile("global_load_async_to_lds_b128 %0, %1, off"
                     :: "v"(lds_off), "v"(ga) : "memory");
        __builtin_amdgcn_s_wait_asynccnt(0);
    }
    __syncthreads();
    const int req = s_req;

    const f4* __restrict__ t4 = (const f4*)trow;
    const f4* __restrict__ d4 = (const f4*)drow;
    const f4* __restrict__ q4 = (const f4*)(q + (size_t)req * (size_t)V);

    float bT = -FLT_MAX; int iT = 0x7FFFFFFF;  // argmax(target)
    float bR = -FLT_MAX; int iR = 0x7FFFFFFF;  // argmax(relu(t-d)/q)

    const int nvec = V >> 2;
    for (int i = tid; i < nvec; i += TPB) {
        // look-ahead into L2 (global_prefetch_b8); speculative, OOB is dropped
        __builtin_prefetch(&t4[i + 2 * TPB], 0, 0);
        __builtin_prefetch(&d4[i + 2 * TPB], 0, 0);
        // single-use streams: non-temporal so the reused q rows keep L2
        f4 t  = __builtin_nontemporal_load(&t4[i]);
        f4 d  = __builtin_nontemporal_load(&d4[i]);
        f4 qq = q4[i];
        const int e = i << 2;
#pragma unroll
        for (int c = 0; c < 4; ++c) {
            float tv = t[c];
            if (tv > bT) { bT = tv; iT = e + c; }
            float r = tv - d[c];
            r = (r > 0.0f) ? r : 0.0f;
            float rv = r / qq[c];              // exact divide to match jnp
            if (rv > bR) { bR = rv; iR = e + c; }
        }
    }
    // generic scalar tail (V % 4)
    for (int e = (nvec << 2) + tid; e < V; e += TPB) {
        float tv = trow[e];
        if (tv > bT) { bT = tv; iT = e; }
        float r = tv - drow[e];
        r = (r > 0.0f) ? r : 0.0f;
        float rv = r / (q + (size_t)req * (size_t)V)[e];
        if (rv > bR) { bR = rv; iR = e; }
    }

    sTv[tid] = bT; sTi[tid] = iT;
    sRv[tid] = bR; sRi[tid] = iR;
    __syncthreads();

    // tree reduction; first-occurrence tie-break (lower index wins on equal)
    for (int s = TPB / 2; s > 0; s >>= 1) {
        if (tid < s) {
            float v = sTv[tid + s]; int ix = sTi[tid + s];
            if (v > sTv[tid] || (v == sTv[tid] && ix < sTi[tid])) { sTv[tid] = v; sTi[tid] = ix; }
            v = sRv[tid + s]; ix = sRi[tid + s];
            if (v > sRv[tid] || (v == sRv[tid] && ix < sRi[tid])) { sRv[tid] = v; sRi[tid] = ix; }
        }
        __syncthreads();
    }
    if (tid == 0) {
        tgt_am[row]    = sTi[0];
        recovered[row] = sRi[0];
    }
}

__global__ __launch_bounds__(64) void rs_finalize(
    const int*   __restrict__ out_init,   // (B, S+1) input output_token_ids
    const int*   __restrict__ cu,         // (B)
    const int*   __restrict__ draft_tok,  // (N)
    const int*   __restrict__ bonus,      // (B)
    const float* __restrict__ uniform,    // (N)
    const unsigned char* __restrict__ is_greedy, // (B) bool
    const int*   __restrict__ tgt_am,     // (N)
    const int*   __restrict__ recovered,  // (N)
    const float* __restrict__ dp,         // (N)
    const float* __restrict__ tp,         // (N)
    int* __restrict__ out,                // (B, S+1)
    int B, int N, int Sl)
{
    const int b = blockIdx.x * blockDim.x + threadIdx.x;
    if (b >= B) return;
    const int Sp1 = Sl + 1;

    const int cub   = cu[b];
    const int cup   = (b > 0) ? cu[b - 1] : 0;
    const int n_per = cub - cup;
    const int start = cub - n_per;
    const bool greedy = (is_greedy[b] != 0);

    int* orow = out + (size_t)b * Sp1;
    const int* irow = out_init + (size_t)b * Sp1;
    for (int j = 0; j < Sp1; ++j) orow[j] = irow[j];

    // first mismatch between draft token and target argmax (greedy path)
    int first_mm = n_per;
    for (int p = 0; p < Sl && p < n_per; ++p) {
        int g = start + p; g = g < 0 ? 0 : (g > N - 1 ? N - 1 : g);
        if (draft_tok[g] != tgt_am[g]) { first_mm = p; break; }
    }

    if (greedy) {
        int copy_len = first_mm + 1; if (copy_len > n_per) copy_len = n_per;
        for (int j = 0; j < Sp1 && j < copy_len; ++j) {
            int g2 = start + j; g2 = g2 < 0 ? 0 : (g2 > N - 1 ? N - 1 : g2);
            orow[j] = tgt_am[g2];
        }
        if (first_mm >= n_per && n_per < Sp1) orow[n_per] = bonus[b];
    } else {
        // first rejection position
        int first_rej = Sl;
        for (int p = 0; p < Sl && p < n_per; ++p) {
            int g = start + p; g = g < 0 ? 0 : (g > N - 1 ? N - 1 : g);
            float dpv = dp[g];
            bool first_acc = (dpv > 0.0f) && (tp[g] / dpv >= uniform[g]);
            if (!first_acc) { first_rej = p; break; }
        }
        // accepted prefix takes draft tokens; rejection point takes recovered
        for (int p = 0; p < Sl && p < n_per; ++p) {
            int g = start + p; g = g < 0 ? 0 : (g > N - 1 ? N - 1 : g);
            if (p < first_rej) {
                orow[p] = draft_tok[g];
            } else if (p == first_rej) {
                orow[p] = recovered[g];
                break;
            } else {
                break;
            }
        }
        if (first_rej >= n_per && n_per < Sp1) orow[n_per] = bonus[b];
    }
}

extern "C" void kernel_launch(void* const* d_in, const int* in_sizes, int n_in,
                              void* d_out, int out_size, void* d_ws, size_t ws_size,
                              hipStream_t stream) {
    const int*   out_init  = (const int*)d_in[0];   // (B, S+1)
    const int*   cu        = (const int*)d_in[1];   // (B)
    const int*   draft_tok = (const int*)d_in[2];   // (N)
    const float* dprob     = (const float*)d_in[3]; // (N, V)
    const float* tprob     = (const float*)d_in[4]; // (N, V)
    const int*   bonus     = (const int*)d_in[5];   // (B)
    const float* uniform   = (const float*)d_in[6]; // (N)
    const float* q         = (const float*)d_in[7]; // (B, V)
    const unsigned char* greedy = (const unsigned char*)d_in[8]; // (B) bool

    const int B   = in_sizes[1];
    const int N   = in_sizes[2];
    const int V   = in_sizes[3] / N;
    const int Sp1 = in_sizes[0] / B;
    const int Sl  = Sp1 - 1;

    int*   tgt_am    = (int*)d_ws;
    int*   recovered = tgt_am + N;
    float* dp        = (float*)(recovered + N);
    float* tp        = dp + N;

    rs_row_scan<<<N, TPB, 0, stream>>>(tprob, dprob, q, cu, draft_tok,
                                       tgt_am, recovered, dp, tp, B, N, V);

    const int fb = 64;
    rs_finalize<<<(B + fb - 1) / fb, fb, 0, stream>>>(
        out_init, cu, draft_tok, bonus, uniform, greedy,
        tgt_am, recovered, dp, tp, (int*)d_out, B, N, Sl);
}